// STTP_57990648431184
// MI455X (gfx1250) — compile-verified
//
#include <hip/hip_runtime.h>
#include <math.h>

#define BB 32
#define DD 128
#define NN 883
#define LL 2
#define RR 18
#define HHD 4
#define DHD 32
#define SSL (BB*LL)              // 64 slices (b,l)
#define TTOK (SSL*NN)            // 56512 tokens
#define TDSZ ((size_t)TTOK*DD)   // 7,233,536 floats
#define KTILES 56                // ceil(883/16)

typedef float v2f __attribute__((ext_vector_type(2)));
typedef float v8f __attribute__((ext_vector_type(8)));

__device__ __forceinline__ v8f wmma4(v2f a, v2f b, v8f c) {
  // V_WMMA_F32_16X16X4_F32 : D = A(16x4,f32) * B(4x16,f32) + C(16x16,f32)
  return __builtin_amdgcn_wmma_f32_16x16x4_f32(false, a, false, b, (short)0, c, false, false);
}

__device__ __forceinline__ v2f ld2(const float* __restrict__ p) {
  const float2 t = *reinterpret_cast<const float2*>(p);
  v2f r; r.x = t.x; r.y = t.y; return r;
}

// ---------------------------------------------------------------------------
// WMMA GEMM: OUT[t, o] = act( sum_k IN[t*inS + k] * W[o*IC + k] + bias[o] )
// act: 0=none 1=relu 2=sigmoid 3=gelu(exact)
// One wave computes a 32x64 tile (8 accumulators): 6 fragment loads feed
// 8 WMMAs per K-step, and the K loop is software-pipelined (next step's
// fragments are loaded into fresh registers before the current WMMAs issue)
// so waits are partial instead of loadcnt==0 per WMMA.
// T multiple of 32, OC multiple of 64, IC multiple of 4.
// ---------------------------------------------------------------------------
__global__ void k_gemm(const float* __restrict__ IN, const float* __restrict__ W,
                       const float* __restrict__ bias, float* __restrict__ OUT,
                       int T, int IC, int OC, int inS, int outS, int colOff, int act)
{
  const int wave = blockIdx.x * (blockDim.x >> 5) + (threadIdx.x >> 5);
  const int lane = threadIdx.x & 31;
  const int colQuads = OC >> 6;          // 64-wide column groups
  const int rowPairs = T >> 5;           // 32-row groups
  const int rp = wave / colQuads;
  if (rp >= rowPairs) return;
  const int cq = wave - rp * colQuads;
  const int m  = lane & 15;
  const int hl = lane >> 4;

  const float* __restrict__ A0 = IN + (size_t)(rp*32 +  0 + m) * inS + 2*hl;
  const float* __restrict__ A1 = IN + (size_t)(rp*32 + 16 + m) * inS + 2*hl;
  const float* __restrict__ B0 = W + (size_t)(cq*64 +  0 + m) * IC + 2*hl;
  const float* __restrict__ B1 = W + (size_t)(cq*64 + 16 + m) * IC + 2*hl;
  const float* __restrict__ B2 = W + (size_t)(cq*64 + 32 + m) * IC + 2*hl;
  const float* __restrict__ B3 = W + (size_t)(cq*64 + 48 + m) * IC + 2*hl;

  v8f acc[8] = {{}, {}, {}, {}, {}, {}, {}, {}};

  // prologue: fragments for k=0
  v2f a0c = ld2(A0), a1c = ld2(A1);
  v2f b0c = ld2(B0), b1c = ld2(B1), b2c = ld2(B2), b3c = ld2(B3);

  int k0 = 0;
  for (; k0 + 4 < IC; k0 += 4) {
    const int kn = k0 + 4;
    // issue next-step loads first (independent registers -> latency hiding)
    v2f a0n = ld2(A0 + kn), a1n = ld2(A1 + kn);
    v2f b0n = ld2(B0 + kn), b1n = ld2(B1 + kn);
    v2f b2n = ld2(B2 + kn), b3n = ld2(B3 + kn);
    acc[0] = wmma4(a0c, b0c, acc[0]);
    acc[1] = wmma4(a0c, b1c, acc[1]);
    acc[2] = wmma4(a0c, b2c, acc[2]);
    acc[3] = wmma4(a0c, b3c, acc[3]);
    acc[4] = wmma4(a1c, b0c, acc[4]);
    acc[5] = wmma4(a1c, b1c, acc[5]);
    acc[6] = wmma4(a1c, b2c, acc[6]);
    acc[7] = wmma4(a1c, b3c, acc[7]);
    a0c = a0n; a1c = a1n;
    b0c = b0n; b1c = b1n; b2c = b2n; b3c = b3n;
  }
  // tail step
  acc[0] = wmma4(a0c, b0c, acc[0]);
  acc[1] = wmma4(a0c, b1c, acc[1]);
  acc[2] = wmma4(a0c, b2c, acc[2]);
  acc[3] = wmma4(a0c, b3c, acc[3]);
  acc[4] = wmma4(a1c, b0c, acc[4]);
  acc[5] = wmma4(a1c, b1c, acc[5]);
  acc[6] = wmma4(a1c, b2c, acc[6]);
  acc[7] = wmma4(a1c, b3c, acc[7]);

#pragma unroll
  for (int rt = 0; rt < 2; ++rt) {
#pragma unroll
    for (int c = 0; c < 4; ++c) {
      const float bvl = bias ? bias[cq*64 + c*16 + m] : 0.0f;
      const int col = colOff + cq*64 + c*16 + m;
#pragma unroll
      for (int j = 0; j < 8; ++j) {
        const int r = rp*32 + rt*16 + hl*8 + j;
        float v = acc[rt*4 + c][j] + bvl;
        if      (act == 1) v = fmaxf(v, 0.0f);
        else if (act == 2) v = 1.0f / (1.0f + expf(-v));
        else if (act == 3) v = 0.5f * v * (1.0f + erff(v * 0.70710678118654752f));
        OUT[(size_t)r * outS + col] = v;
      }
    }
  }
}

// ---------------------------------------------------------------------------
// Flash neighbor attention with post-softmax multiplicative mask (simi), WMMA.
// grid (56 qtiles, 4 heads, 64 slices), block = 32 (one wave).
// Q/K/V: [TTOK,128] token-major; head h uses cols h*32..h*32+31. O same layout.
// o[q,:] = (1/Z_q) * sum_k exp(e_qk - m_q) * simi[q,k] * v[k],  Z without mask.
// ---------------------------------------------------------------------------
__global__ void k_flash(const float* __restrict__ Q, const float* __restrict__ K,
                        const float* __restrict__ V, const float* __restrict__ simi,
                        float* __restrict__ O)
{
  const int qt = blockIdx.x, h = blockIdx.y, s = blockIdx.z;
  const int lane = threadIdx.x;
  const int m  = lane & 15;
  const int hl = lane >> 4;
  const float scale = 0.17677669529663688f;  // 1/sqrt(32)
  const size_t base = (size_t)s * NN * DD + h * DHD;

  __shared__ float lp[16 * 20];

  const int q  = qt*16 + m;
  const int qc = (q < NN) ? q : (NN-1);
  const float* qrow = Q + base + (size_t)qc * DD + 2*hl;
  v2f qf[8];
#pragma unroll
  for (int i = 0; i < 8; ++i) {
    v2f t = ld2(qrow + 4*i);
    qf[i].x = t.x * scale; qf[i].y = t.y * scale;
  }

  v8f acc0 = {}, acc1 = {};
  float rm[8], rz[8];
#pragma unroll
  for (int j = 0; j < 8; ++j) { rm[j] = -__builtin_inff(); rz[j] = 0.0f; }

  for (int kt = 0; kt < KTILES; ++kt) {
    const int kn  = kt*16 + m;
    const int knc = (kn < NN) ? kn : (NN-1);
    const float* krow = K + base + (size_t)knc * DD + 2*hl;
    // load all K fragments into independent registers, then run the 8 WMMAs
    v2f kf[8];
#pragma unroll
    for (int i = 0; i < 8; ++i) kf[i] = ld2(krow + 4*i);
    v8f sacc = {};
#pragma unroll
    for (int i = 0; i < 8; ++i) sacc = wmma4(qf[i], kf[i], sacc);

    // issue V fragment loads now; their latency hides under the softmax VALU
    v2f vb0[4], vb1[4];
#pragma unroll
    for (int i = 0; i < 4; ++i) {
      int kr0 = kt*16 + 4*i + 2*hl;
      int kr1 = kr0 + 1;
      if (kr0 >= NN) kr0 = NN-1;
      if (kr1 >= NN) kr1 = NN-1;
      const float* v0 = V + base + (size_t)kr0 * DD;
      const float* v1 = V + base + (size_t)kr1 * DD;
      vb0[i].x = v0[m];      vb0[i].y = v1[m];
      vb1[i].x = v0[16 + m]; vb1[i].y = v1[16 + m];
    }

    const bool oob = (kn >= NN);
    const float* srow = simi + (size_t)knc;   // + qg*NN below
#pragma unroll
    for (int j = 0; j < 8; ++j) {
      float sv = oob ? -__builtin_inff() : sacc[j];
      float red = sv;
      red = fmaxf(red, __shfl_xor(red, 1, 32));
      red = fmaxf(red, __shfl_xor(red, 2, 32));
      red = fmaxf(red, __shfl_xor(red, 4, 32));
      red = fmaxf(red, __shfl_xor(red, 8, 32));
      const float mnew  = fmaxf(rm[j], red);
      const float alpha = expf(rm[j] - mnew);
      rm[j] = mnew;
      const float p = expf(sv - mnew);        // 0 for OOB keys
      float su = p;
      su += __shfl_xor(su, 1, 32);
      su += __shfl_xor(su, 2, 32);
      su += __shfl_xor(su, 4, 32);
      su += __shfl_xor(su, 8, 32);
      rz[j] = rz[j] * alpha + su;             // Z excludes mask (post-softmax mask)
      acc0[j] *= alpha; acc1[j] *= alpha;
      int qg = qt*16 + hl*8 + j; if (qg >= NN) qg = NN-1;
      lp[(hl*8 + j)*20 + m] = p * srow[(size_t)qg * NN];
    }
    __syncthreads();
#pragma unroll
    for (int i = 0; i < 4; ++i) {
      v2f a;
      a.x = lp[m*20 + 4*i + 2*hl];
      a.y = lp[m*20 + 4*i + 2*hl + 1];
      acc0 = wmma4(a, vb0[i], acc0);
      acc1 = wmma4(a, vb1[i], acc1);
    }
    __syncthreads();
  }
#pragma unroll
  for (int j = 0; j < 8; ++j) {
    const int qg = qt*16 + hl*8 + j;
    if (qg < NN) {
      const float inv = 1.0f / rz[j];
      O[base + (size_t)qg * DD + m]      = acc0[j] * inv;
      O[base + (size_t)qg * DD + 16 + m] = acc1[j] * inv;
    }
  }
}

// ---------------------------------------------------------------------------
// Per-batch LayerNorm stats over D*N*L: stats[2b]=mean, stats[2b+1]=rstd.
// ---------------------------------------------------------------------------
__global__ void k_stats(const float* __restrict__ X, float* __restrict__ stats)
{
  const int b = blockIdx.x;
  const size_t cnt = (size_t)DD*NN*LL;
  const float* p = X + (size_t)b * cnt;
  double s = 0.0, s2 = 0.0;
  for (size_t i = threadIdx.x; i < cnt; i += blockDim.x) {
    const float v = p[i]; s += v; s2 += (double)v * v;
  }
  __shared__ double sh[256], sh2[256];
  sh[threadIdx.x] = s; sh2[threadIdx.x] = s2;
  __syncthreads();
  for (int off = 128; off > 0; off >>= 1) {
    if ((int)threadIdx.x < off) { sh[threadIdx.x] += sh[threadIdx.x+off]; sh2[threadIdx.x] += sh2[threadIdx.x+off]; }
    __syncthreads();
  }
  if (threadIdx.x == 0) {
    const double mean = sh[0] / (double)cnt;
    const double var  = sh2[0] / (double)cnt - mean*mean;
    stats[2*b]   = (float)mean;
    stats[2*b+1] = (float)(1.0 / sqrt(var + 1e-6));
  }
}

// [B,D,N,L] -> token-major [T,128] with SiLU (for adaLN input).
__global__ void k_pack_silu(const float* __restrict__ XM, float* __restrict__ P)
{
  __shared__ float t[16][17];
  const int nT = blockIdx.x, dT = blockIdx.y, s = blockIdx.z;
  const int b = s >> 1, l = s & 1;
  const int tx = threadIdx.x, ty = threadIdx.y;
  const int d = dT*16 + ty, n = nT*16 + tx;
  if (n < NN) {
    const float v = XM[(((size_t)b*DD + d)*NN + n)*LL + l];
    t[ty][tx] = v / (1.0f + expf(-v));
  }
  __syncthreads();
  const int d2 = dT*16 + tx, n2 = nT*16 + ty;
  if (n2 < NN) P[((size_t)s*NN + n2)*DD + d2] = t[tx][ty];
}

// h = (LN(x)) * (1 + ada[scOff]) + ada[shOff]  -> token-major
__global__ void k_modulate(const float* __restrict__ X, const float* __restrict__ LW,
                           const float* __restrict__ LB, const float* __restrict__ ADA,
                           const float* __restrict__ STATS, float* __restrict__ OUT,
                           int shOff, int scOff)
{
  __shared__ float tv[16][17];
  const int nT = blockIdx.x, dT = blockIdx.y, s = blockIdx.z;
  const int b = s >> 1, l = s & 1;
  const int tx = threadIdx.x, ty = threadIdx.y;
  const float mu = STATS[2*b], rstd = STATS[2*b+1];
  const int d = dT*16 + ty, n = nT*16 + tx;
  if (n < NN) {
    const size_t wo = ((size_t)d*NN + n)*LL + l;
    const float v = X[(((size_t)b*DD + d)*NN + n)*LL + l];
    tv[ty][tx] = (v - mu) * rstd * LW[wo] + LB[wo];
  }
  __syncthreads();
  const int d2 = dT*16 + tx, n2 = nT*16 + ty;
  if (n2 < NN) {
    const size_t tk = (size_t)s*NN + n2;
    const size_t ab = tk * 768;
    OUT[tk*DD + d2] = tv[tx][ty] * (1.0f + ADA[ab + scOff + d2]) + ADA[ab + shOff + d2];
  }
}

// x2 = x + g_msa * ( alp*rf + (1-alp)*nf )   (x-layout output)
__global__ void k_mix(const float* __restrict__ X, const float* __restrict__ ADA,
                      const float* __restrict__ CON, const float* __restrict__ ALP,
                      float* __restrict__ X2)
{
  __shared__ float t[16][17];
  const int nT = blockIdx.x, dT = blockIdx.y, s = blockIdx.z;
  const int b = s >> 1, l = s & 1;
  const int tx = threadIdx.x, ty = threadIdx.y;
  const int d = dT*16 + tx, n = nT*16 + ty;
  if (n < NN) {
    const size_t tk = (size_t)s*NN + n;
    const float rf = CON[tk*256 + d];
    const float nf = CON[tk*256 + 128 + d];
    const float al = ALP[tk*DD + d];
    t[ty][tx] = ADA[tk*768 + 256 + d] * (al*rf + (1.0f - al)*nf);
  }
  __syncthreads();
  const int d2 = dT*16 + ty, n2 = nT*16 + tx;
  if (n2 < NN) {
    const size_t off = (((size_t)b*DD + d2)*NN + n2)*LL + l;
    X2[off] = X[off] + t[tx][ty];
  }
}

// out = x2 + g_mlp * y2   (x-layout output)
__global__ void k_out(const float* __restrict__ X2, const float* __restrict__ ADA,
                      const float* __restrict__ Y2, float* __restrict__ OUT)
{
  __shared__ float t[16][17];
  const int nT = blockIdx.x, dT = blockIdx.y, s = blockIdx.z;
  const int b = s >> 1, l = s & 1;
  const int tx = threadIdx.x, ty = threadIdx.y;
  const int d = dT*16 + tx, n = nT*16 + ty;
  if (n < NN) {
    const size_t tk = (size_t)s*NN + n;
    t[ty][tx] = ADA[tk*768 + 640 + d] * Y2[tk*DD + d];
  }
  __syncthreads();
  const int d2 = dT*16 + ty, n2 = nT*16 + tx;
  if (n2 < NN) {
    const size_t off = (((size_t)b*DD + d2)*NN + n2)*LL + l;
    OUT[off] = X2[off] + t[tx][ty];
  }
}

// softmax(ass) rows + row norms
__global__ void k_asoft(const float* __restrict__ ASS, float* __restrict__ A,
                        float* __restrict__ NRM)
{
  const int n = blockIdx.x * blockDim.x + threadIdx.x;
  if (n >= NN) return;
  float mx = -__builtin_inff();
  for (int r = 0; r < RR; ++r) mx = fmaxf(mx, ASS[n*RR + r]);
  float e[RR]; float sum = 0.0f;
  for (int r = 0; r < RR; ++r) { e[r] = expf(ASS[n*RR + r] - mx); sum += e[r]; }
  const float inv = 1.0f / sum;
  float sq = 0.0f;
  for (int r = 0; r < RR; ++r) { const float v = e[r]*inv; A[n*RR + r] = v; sq += v*v; }
  NRM[n] = fmaxf(sqrtf(sq), 1e-8f);
}

// cosine similarity matrix
__global__ void k_simi(const float* __restrict__ A, const float* __restrict__ NRM,
                       float* __restrict__ SIMI)
{
  const int idx = blockIdx.x * blockDim.x + threadIdx.x;
  if (idx >= NN*NN) return;
  const int i = idx / NN, j = idx - i*NN;
  float dot = 0.0f;
  for (int r = 0; r < RR; ++r) dot += A[i*RR + r] * A[j*RR + r];
  SIMI[idx] = dot / (NRM[i] * NRM[j]);
}

// rf0[s,r,d] = sum_n a[n,r] * H[s*N+n, d]
__global__ void k_rf0(const float* __restrict__ A, const float* __restrict__ H,
                      float* __restrict__ RF0)
{
  const int r = blockIdx.x, s = blockIdx.y, d = threadIdx.x;
  float acc = 0.0f;
  for (int n = 0; n < NN; ++n)
    acc += A[n*RR + r] * H[((size_t)s*NN + n)*DD + d];
  RF0[((size_t)s*RR + r)*DD + d] = acc;
}

// tiny R=18 MHA (no mask): one thread per (s,h,q)
__global__ void k_smallattn(const float* __restrict__ Q, const float* __restrict__ K,
                            const float* __restrict__ V, float* __restrict__ O)
{
  const int idx = blockIdx.x * blockDim.x + threadIdx.x;
  if (idx >= SSL*HHD*RR) return;
  const int q = idx % RR;
  const int h = (idx / RR) % HHD;
  const int s = idx / (RR*HHD);
  const float scale = 0.17677669529663688f;
  const float* Qp = Q + ((size_t)s*RR + q)*DD + h*DHD;
  float e[RR]; float mx = -__builtin_inff();
  for (int k = 0; k < RR; ++k) {
    const float* Kp = K + ((size_t)s*RR + k)*DD + h*DHD;
    float dv = 0.0f;
    for (int dd = 0; dd < DHD; ++dd) dv += Qp[dd] * Kp[dd];
    e[k] = dv * scale; mx = fmaxf(mx, e[k]);
  }
  float sum = 0.0f;
  for (int k = 0; k < RR; ++k) { e[k] = expf(e[k] - mx); sum += e[k]; }
  const float inv = 1.0f / sum;
  float out[DHD];
  for (int dd = 0; dd < DHD; ++dd) out[dd] = 0.0f;
  for (int k = 0; k < RR; ++k) {
    const float w = e[k] * inv;
    const float* Vp = V + ((size_t)s*RR + k)*DD + h*DHD;
    for (int dd = 0; dd < DHD; ++dd) out[dd] += w * Vp[dd];
  }
  float* Op = O + ((size_t)s*RR + q)*DD + h*DHD;
  for (int dd = 0; dd < DHD; ++dd) Op[dd] = out[dd];
}

// CON[t, 0:128] = sum_r a[n,r] * RFO[s*18+r, d]
__global__ void k_rf_expand(const float* __restrict__ A, const float* __restrict__ RFO,
                            float* __restrict__ CON)
{
  const size_t idx = (size_t)blockIdx.x * blockDim.x + threadIdx.x;
  if (idx >= TDSZ) return;
  const int d = (int)(idx & 127);
  const size_t tk = idx >> 7;
  const int n = (int)(tk % NN);
  const int s = (int)(tk / NN);
  float acc = 0.0f;
  for (int r = 0; r < RR; ++r)
    acc += A[n*RR + r] * RFO[((size_t)s*RR + r)*DD + d];
  CON[tk*256 + d] = acc;
}

// ---------------------------------------------------------------------------
static void gemm(hipStream_t st, const float* IN, const float* W, const float* b,
                 float* OUT, int T, int IC, int OC, int inS, int outS, int colOff, int act)
{
  const long tiles = (long)(T/32) * (OC/64);
  const int blocks = (int)((tiles + 7) / 8);
  k_gemm<<<blocks, 256, 0, st>>>(IN, W, b, OUT, T, IC, OC, inS, outS, colOff, act);
}

extern "C" void kernel_launch(void* const* d_in, const int* in_sizes, int n_in,
                              void* d_out, int out_size, void* d_ws, size_t ws_size,
                              hipStream_t stream)
{
  (void)in_sizes; (void)n_in; (void)out_size; (void)ws_size;
  const float* x       = (const float*)d_in[0];
  const float* x_mark  = (const float*)d_in[1];
  const float* adaln_w = (const float*)d_in[2];
  const float* adaln_b = (const float*)d_in[3];
  const float* ln1_w   = (const float*)d_in[4];
  const float* ln1_b   = (const float*)d_in[5];
  const float* ln2_w   = (const float*)d_in[6];
  const float* ln2_b   = (const float*)d_in[7];
  const float* ass     = (const float*)d_in[8];
  const float* ra_qw = (const float*)d_in[9];  const float* ra_qb = (const float*)d_in[10];
  const float* ra_kw = (const float*)d_in[11]; const float* ra_kb = (const float*)d_in[12];
  const float* ra_vw = (const float*)d_in[13]; const float* ra_vb = (const float*)d_in[14];
  const float* ra_pw = (const float*)d_in[15]; const float* ra_pb = (const float*)d_in[16];
  const float* na_qw = (const float*)d_in[17]; const float* na_qb = (const float*)d_in[18];
  const float* na_kw = (const float*)d_in[19]; const float* na_kb = (const float*)d_in[20];
  const float* na_vw = (const float*)d_in[21]; const float* na_vb = (const float*)d_in[22];
  const float* na_pw = (const float*)d_in[23]; const float* na_pb = (const float*)d_in[24];
  const float* a1_w  = (const float*)d_in[25]; const float* a1_b  = (const float*)d_in[26];
  const float* a2_w  = (const float*)d_in[27]; const float* a2_b  = (const float*)d_in[28];
  const float* m1_w  = (const float*)d_in[29]; const float* m1_b  = (const float*)d_in[30];
  const float* m2_w  = (const float*)d_in[31]; const float* m2_b  = (const float*)d_in[32];

  float* ws = (float*)d_ws;
  size_t off = 0;
  float* ADA  = ws + off; off += (size_t)TTOK * 768;   // all 6 adaLN chunks
  float* Hbuf = ws + off; off += TDSZ;                 // h (later h2)
  float* CON  = ws + off; off += (size_t)TTOK * 256;   // [rf | nf]
  float* BIG  = ws + off; off += 4 * TDSZ;             // Q|K|V|NFO -> G1 -> Y1|Y2
  float* ALP  = ws + off; off += TDSZ;
  float* X2   = ws + off; off += TDSZ;
  float* Pk   = ws + off; off += TDSZ;                 // silu(x_mark) packed
  float* ASOFT= ws + off; off += (size_t)NN * RR;
  float* NRM  = ws + off; off += NN;
  float* SIMI = ws + off; off += (size_t)NN * NN;
  float* RF0  = ws + off; off += (size_t)SSL * RR * DD;
  float* RQ   = ws + off; off += (size_t)SSL * RR * DD;
  float* RK   = ws + off; off += (size_t)SSL * RR * DD;
  float* RV   = ws + off; off += (size_t)SSL * RR * DD;
  float* O1R  = ws + off; off += (size_t)SSL * RR * DD;
  float* RFO  = ws + off; off += (size_t)SSL * RR * DD;
  float* ST1  = ws + off; off += 2 * BB;
  float* ST2  = ws + off; off += 2 * BB;

  const dim3 tb(16, 16);
  const dim3 tg(KTILES, DD/16, SSL);                   // transpose-style grids

  // LN1 stats + adaLN modulation table
  k_stats<<<BB, 256, 0, stream>>>(x, ST1);
  k_pack_silu<<<tg, tb, 0, stream>>>(x_mark, Pk);
  gemm(stream, Pk, adaln_w, adaln_b, ADA, TTOK, 128, 768, 128, 768, 0, 0);
  k_modulate<<<tg, tb, 0, stream>>>(x, ln1_w, ln1_b, ADA, ST1, Hbuf, 0, 128);

  // soft assignment, similarity
  k_asoft<<<(NN + 127)/128, 128, 0, stream>>>(ass, ASOFT, NRM);
  k_simi<<<(NN*NN + 255)/256, 256, 0, stream>>>(ASOFT, NRM, SIMI);

  // regional attention (R=18)
  k_rf0<<<dim3(RR, SSL), 128, 0, stream>>>(ASOFT, Hbuf, RF0);
  gemm(stream, RF0, ra_qw, ra_qb, RQ, SSL*RR, 128, 128, 128, 128, 0, 0);
  gemm(stream, RF0, ra_kw, ra_kb, RK, SSL*RR, 128, 128, 128, 128, 0, 0);
  gemm(stream, RF0, ra_vw, ra_vb, RV, SSL*RR, 128, 128, 128, 128, 0, 0);
  k_smallattn<<<(SSL*HHD*RR + 127)/128, 128, 0, stream>>>(RQ, RK, RV, O1R);
  gemm(stream, O1R, ra_pw, ra_pb, RFO, SSL*RR, 128, 128, 128, 128, 0, 0);
  k_rf_expand<<<(int)((TDSZ + 255)/256), 256, 0, stream>>>(ASOFT, RFO, CON);

  // neighbor attention (N=883, masked post-softmax) with flash WMMA
  gemm(stream, Hbuf, na_qw, na_qb, BIG,          TTOK, 128, 128, 128, 128, 0, 0);
  gemm(stream, Hbuf, na_kw, na_kb, BIG + TDSZ,   TTOK, 128, 128, 128, 128, 0, 0);
  gemm(stream, Hbuf, na_vw, na_vb, BIG + 2*TDSZ, TTOK, 128, 128, 128, 128, 0, 0);
  k_flash<<<dim3(KTILES, HHD, SSL), 32, 0, stream>>>(BIG, BIG + TDSZ, BIG + 2*TDSZ,
                                                     SIMI, BIG + 3*TDSZ);
  gemm(stream, BIG + 3*TDSZ, na_pw, na_pb, CON, TTOK, 128, 128, 128, 256, 128, 0);

  // gating MLP (reuses BIG as the 512-wide hidden)
  gemm(stream, CON, a1_w, a1_b, BIG, TTOK, 256, 512, 256, 512, 0, 1);
  gemm(stream, BIG, a2_w, a2_b, ALP, TTOK, 512, 128, 512, 128, 0, 2);

  // residual 1
  k_mix<<<tg, tb, 0, stream>>>(x, ADA, CON, ALP, X2);

  // LN2 + MLP + residual 2
  k_stats<<<BB, 256, 0, stream>>>(X2, ST2);
  k_modulate<<<tg, tb, 0, stream>>>(X2, ln2_w, ln2_b, ADA, ST2, Hbuf, 384, 512);
  gemm(stream, Hbuf, m1_w, m1_b, BIG,        TTOK, 128, 128, 128, 128, 0, 3);
  gemm(stream, BIG,  m2_w, m2_b, BIG + TDSZ, TTOK, 128, 128, 128, 128, 0, 0);
  k_out<<<tg, tb, 0, stream>>>(X2, ADA, BIG + TDSZ, (float*)d_out);
}